// NegativeSamplingModel_41480794145350
// MI455X (gfx1250) — compile-verified
//
#include <hip/hip_runtime.h>
#include <hip/hip_bf16.h>

// NegativeSampling dot: out[b] = dot(W_word[idx[b,1]], W_ctx[idx[b,0]]), D=32.
// Memory-bound gather (AI ~ 0.23 FLOP/B); tables are L2-resident (25.6 MB << 192 MB).
// Compute is routed through V_WMMA_F32_16X16X4_F32: 16 batch elements per wave,
// 8 accumulating WMMAs over K=32, results on the accumulator diagonal.
// Per-lane float2 global loads place gathered rows directly into the WMMA
// A/B fragment layout (no LDS, no transpose, no shuffles).

typedef __attribute__((ext_vector_type(2))) float v2f;
typedef __attribute__((ext_vector_type(8))) float v8f;

#define EMBED_DIM 32
#define ELEMS_PER_WAVE 16
#define BLOCK_THREADS 256
#define WAVES_PER_BLOCK (BLOCK_THREADS / 32)
#define ELEMS_PER_BLOCK (WAVES_PER_BLOCK * ELEMS_PER_WAVE)

__global__ __launch_bounds__(BLOCK_THREADS) void nsdot_wmma_f32_kernel(
    const int* __restrict__ inputs,    // [BATCH, 2] int32: (ctx_idx, word_idx)
    const float* __restrict__ W_word,  // [VOCAB, 32] fp32
    const float* __restrict__ W_ctx,   // [VOCAB, 32] fp32
    float* __restrict__ out,           // [BATCH] fp32
    int batch)
{
    const int lane = threadIdx.x & 31;
    const int wave = threadIdx.x >> 5;
    const int gwave = blockIdx.x * WAVES_PER_BLOCK + wave;
    const int base = gwave * ELEMS_PER_WAVE;
    if (base >= batch) return;  // wave-uniform: EXEC stays all-ones below

    const int m  = lane & 15;   // which batch element of the 16 (M for A, N for B)
    const int hi = lane >> 4;   // 0 -> K pair {0,1}, 1 -> K pair {2,3}

    // Defensive clamp (BATCH is a multiple of 16 so this is a no-op in practice);
    // keeps all 32 lanes active through the WMMA loop.
    int b = base + m;
    b = (b < batch) ? b : (batch - 1);

    const int ctx_row  = inputs[2 * b + 0];
    const int word_row = inputs[2 * b + 1];

    // Each lane's slice of the A (word) and B (ctx) fragments is a contiguous
    // float2 at row_base + 4*chunk + 2*hi -> per-lane global_load_b64.
    const float* wp = W_word + (size_t)word_row * EMBED_DIM + 2 * hi;
    const float* cp = W_ctx  + (size_t)ctx_row  * EMBED_DIM + 2 * hi;

    v8f acc = {};
#pragma unroll
    for (int c = 0; c < EMBED_DIM / 4; ++c) {
        v2f a  = *(const v2f*)(wp + 4 * c);  // A: 16x4 fp32 fragment slice
        v2f bm = *(const v2f*)(cp + 4 * c);  // B: 4x16 fp32 fragment slice
        // 8 args: (neg_a, A, neg_b, B, c_mod, C, reuse_a, reuse_b)
        acc = __builtin_amdgcn_wmma_f32_16x16x4_f32(
            false, a, false, bm, (short)0, acc, false, false);
    }

    // Diagonal of the 16x16 f32 accumulator:
    //   i < 8  : lane i,     VGPR i
    //   i >= 8 : lane i+16,  VGPR i-8   (i.e. lanes 24..31 hold outputs 8..15)
    float val = 0.0f;
    int oidx = -1;
#pragma unroll
    for (int r = 0; r < 8; ++r) {
        if (lane == r)      { val = acc[r]; oidx = base + r;     }
        if (lane == r + 24) { val = acc[r]; oidx = base + 8 + r; }
    }
    if (oidx >= 0 && oidx < batch) out[oidx] = val;
}

extern "C" void kernel_launch(void* const* d_in, const int* in_sizes, int n_in,
                              void* d_out, int out_size, void* d_ws, size_t ws_size,
                              hipStream_t stream) {
    const int*   inputs = (const int*)d_in[0];    // [BATCH,2] int32
    const float* W_word = (const float*)d_in[1];  // [VOCAB,32] fp32
    const float* W_ctx  = (const float*)d_in[2];  // [VOCAB,32] fp32
    float*       out    = (float*)d_out;          // [BATCH] fp32

    const int batch = in_sizes[0] / 2;
    const int grid  = (batch + ELEMS_PER_BLOCK - 1) / ELEMS_PER_BLOCK;

    nsdot_wmma_f32_kernel<<<grid, BLOCK_THREADS, 0, stream>>>(
        inputs, W_word, W_ctx, out, batch);
}